// AMG_encoder_23424751632876
// MI455X (gfx1250) — compile-verified
//
#include <hip/hip_runtime.h>

#define NGRIDS 32
#define NFEAT  4
#define GS     64
#define GCELLS (GS*GS*GS)   // 262144 voxels per channel

typedef float v2f __attribute__((ext_vector_type(2)));
typedef float v8f __attribute__((ext_vector_type(8)));

__device__ __forceinline__ int iclamp63(int v) {
    v = v < 0 ? 0 : v;
    return v > (GS - 1) ? (GS - 1) : v;
}

// Build the 3x3 linear part + translation of the per-grid TRS matrix,
// exactly matching the reference build_transform().
__device__ __forceinline__ void compute_transform(const float* __restrict__ sc,
                                                  const float* __restrict__ tr,
                                                  const float* __restrict__ ro,
                                                  int g, float* __restrict__ T) {
    float r0 = ro[g * 3 + 0], r1 = ro[g * 3 + 1], r2 = ro[g * 3 + 2];
    float c0 = cosf(r0), c1 = cosf(r1), c2 = cosf(r2);
    float s0 = sinf(r0), s1 = sinf(r1), s2 = sinf(r2);
    float sx = sc[g * 3 + 0], sy = sc[g * 3 + 1], sz = sc[g * 3 + 2];
    T[0] = sx * c1 * c2;  T[1] = sy * (s0 * s1 * c2 - c0 * s2);  T[2] = sz * (c0 * s1 * c2 + s0 * s2);
    T[3] = sx * c1 * s2;  T[4] = sy * (s0 * s1 * s2 + c0 * c2);  T[5] = sz * (c0 * s1 * s2 - s0 * c2);
    T[6] = -sx * s1;      T[7] = sy * s0 * c1;                   T[8] = sz * c0 * c1;
    T[9]  = tr[g * 3 + 0];
    T[10] = tr[g * 3 + 1];
    T[11] = tr[g * 3 + 2];
}

// Trilinear sample (align_corners=True, zero padding) from channel-last
// repacked grid ws[G][D][H][W][4]: one float4 per corner.
__device__ __forceinline__ float4 sample_cl(const float* __restrict__ ws, int g,
                                            float cx, float cy, float cz) {
    float ix = (cx + 1.0f) * 0.5f * (float)(GS - 1);
    float iy = (cy + 1.0f) * 0.5f * (float)(GS - 1);
    float iz = (cz + 1.0f) * 0.5f * (float)(GS - 1);
    float xf = floorf(ix), yf = floorf(iy), zf = floorf(iz);
    float fx = ix - xf, fy = iy - yf, fz = iz - zf;
    int x0 = (int)xf, y0 = (int)yf, z0 = (int)zf;
    float4 acc; acc.x = 0.f; acc.y = 0.f; acc.z = 0.f; acc.w = 0.f;
#pragma unroll
    for (int dz = 0; dz < 2; ++dz) {
        int zi = z0 + dz; float wz = dz ? fz : 1.0f - fz;
#pragma unroll
        for (int dy = 0; dy < 2; ++dy) {
            int yi = y0 + dy; float wy = dy ? fy : 1.0f - fy;
#pragma unroll
            for (int dx = 0; dx < 2; ++dx) {
                int xi = x0 + dx; float wx = dx ? fx : 1.0f - fx;
                bool valid = ((unsigned)xi < (unsigned)GS) &&
                             ((unsigned)yi < (unsigned)GS) &&
                             ((unsigned)zi < (unsigned)GS);
                int xc = iclamp63(xi), yc = iclamp63(yi), zc = iclamp63(zi);
                size_t vox = (((size_t)(g * GS + zc) * GS + yc) * GS + xc);
                const float4 v = *(const float4*)(ws + vox * 4);
                float w = valid ? (wx * wy * wz) : 0.0f;
                acc.x += w * v.x; acc.y += w * v.y; acc.z += w * v.z; acc.w += w * v.w;
            }
        }
    }
    return acc;
}

// Fallback: sample directly from the original [G][C][D][H][W] layout.
__device__ __forceinline__ float4 sample_orig(const float* __restrict__ feats, int g,
                                              float cx, float cy, float cz) {
    float ix = (cx + 1.0f) * 0.5f * (float)(GS - 1);
    float iy = (cy + 1.0f) * 0.5f * (float)(GS - 1);
    float iz = (cz + 1.0f) * 0.5f * (float)(GS - 1);
    float xf = floorf(ix), yf = floorf(iy), zf = floorf(iz);
    float fx = ix - xf, fy = iy - yf, fz = iz - zf;
    int x0 = (int)xf, y0 = (int)yf, z0 = (int)zf;
    size_t base = (size_t)g * NFEAT * GCELLS;
    float4 acc; acc.x = 0.f; acc.y = 0.f; acc.z = 0.f; acc.w = 0.f;
#pragma unroll
    for (int dz = 0; dz < 2; ++dz) {
        int zi = z0 + dz; float wz = dz ? fz : 1.0f - fz;
#pragma unroll
        for (int dy = 0; dy < 2; ++dy) {
            int yi = y0 + dy; float wy = dy ? fy : 1.0f - fy;
#pragma unroll
            for (int dx = 0; dx < 2; ++dx) {
                int xi = x0 + dx; float wx = dx ? fx : 1.0f - fx;
                bool valid = ((unsigned)xi < (unsigned)GS) &&
                             ((unsigned)yi < (unsigned)GS) &&
                             ((unsigned)zi < (unsigned)GS);
                int xc = iclamp63(xi), yc = iclamp63(yi), zc = iclamp63(zi);
                int off = (zc * GS + yc) * GS + xc;
                float w = valid ? (wx * wy * wz) : 0.0f;
                acc.x += w * feats[base + 0 * GCELLS + off];
                acc.y += w * feats[base + 1 * GCELLS + off];
                acc.z += w * feats[base + 2 * GCELLS + off];
                acc.w += w * feats[base + 3 * GCELLS + off];
            }
        }
    }
    return acc;
}

// Repack [G][C][D][H][W] -> [G][D][H][W][C] so each trilinear corner is one float4.
__global__ __launch_bounds__(256) void amg_repack(const float* __restrict__ in,
                                                  float* __restrict__ ws) {
    size_t v = (size_t)blockIdx.x * 256 + threadIdx.x;   // voxel id, 0..G*GCELLS-1
    int g = (int)(v >> 18);
    int r = (int)(v & (GCELLS - 1));
    size_t base = (size_t)g * NFEAT * GCELLS;
    float4 o;
    o.x = in[base + 0 * GCELLS + r];
    o.y = in[base + 1 * GCELLS + r];
    o.z = in[base + 2 * GCELLS + r];
    o.w = in[base + 3 * GCELLS + r];
    *(float4*)(ws + v * 4) = o;
}

// Main kernel. 256 threads = 8 wave32; each wave owns 32 consecutive points.
// Coordinate transform done with V_WMMA_F32_16X16X4_F32:
//   A (16x4) = 16 points x (x,y,z,1); B (4x16) = 5 grids x 3 coord columns
//   (rows 0..2 = A_g rows transposed, row 3 = translation).
template <bool CL>
__global__ __launch_bounds__(256) void amg_main(const float* __restrict__ x,
                                                const float* __restrict__ sc,
                                                const float* __restrict__ tr,
                                                const float* __restrict__ ro,
                                                const float* __restrict__ feats,
                                                const float* __restrict__ ws,
                                                float* __restrict__ out) {
    __shared__ float T[NGRIDS][12];
    __shared__ float cbuf[8][32][17];   // [wave][point-in-wave][coord column], padded

    const int tid  = threadIdx.x;
    if (tid < NGRIDS) compute_transform(sc, tr, ro, tid, &T[tid][0]);
    __syncthreads();

    const int wave = tid >> 5;
    const int lane = tid & 31;
    const int P = blockIdx.x * 256 + wave * 32;   // wave's base point
    const int q = P + lane;                       // this thread's point

    // A matrices for both 16-point halves of the wave (16x4 fp32 layout:
    // lanes 0-15 hold K=0,1 = (x,y); lanes 16-31 hold K=2,3 = (z,1)).
    v2f a0, a1;
    {
        const int l16 = lane & 15;
        const int q0 = P + l16;
        const int q1 = P + 16 + l16;
        if (lane < 16) {
            a0.x = x[q0 * 3 + 0]; a0.y = x[q0 * 3 + 1];
            a1.x = x[q1 * 3 + 0]; a1.y = x[q1 * 3 + 1];
        } else {
            a0.x = x[q0 * 3 + 2]; a0.y = 1.0f;
            a1.x = x[q1 * 3 + 2]; a1.y = 1.0f;
        }
    }

    float* outrow = out + (size_t)q * (NGRIDS * NFEAT);

    for (int j = 0; j < 7; ++j) {          // 7 groups of up to 5 grids
        // Build B (4x16 fp32: VGPR0 = rows K0|K2, VGPR1 = rows K1|K3).
        const int n  = lane & 15;
        const int jl = n / 3;
        const int d  = n - jl * 3;
        const int g  = j * 5 + jl;
        const bool colok = (n < 15) && (g < NGRIDS);
        const int gg = (g < NGRIDS) ? g : (NGRIDS - 1);
        float bx = (lane < 16) ? T[gg][d * 3 + 0] : T[gg][d * 3 + 2];
        float by = (lane < 16) ? T[gg][d * 3 + 1] : T[gg][9 + d];
        v2f b;
        b.x = colok ? bx : 0.0f;
        b.y = colok ? by : 0.0f;

        v8f cz8 = {};
        v8f d0 = __builtin_amdgcn_wmma_f32_16x16x4_f32(false, a0, false, b,
                                                       (short)0, cz8, false, false);
        v8f d1 = __builtin_amdgcn_wmma_f32_16x16x4_f32(false, a1, false, b,
                                                       (short)0, cz8, false, false);

        // Scatter D (16x16 f32 layout: VGPR r -> row r (lanes 0-15) / r+8 (16-31)).
        const int colw  = lane & 15;
        const int rbase = (lane < 16) ? 0 : 8;
#pragma unroll
        for (int r = 0; r < 8; ++r) {
            cbuf[wave][rbase + r][colw]      = d0[r];
            cbuf[wave][16 + rbase + r][colw] = d1[r];
        }
        __syncthreads();

        // Each thread samples the group's grids for its own point.
#pragma unroll
        for (int jl2 = 0; jl2 < 5; ++jl2) {
            const int g2 = j * 5 + jl2;
            if (g2 >= NGRIDS) break;       // uniform branch
            float cx = cbuf[wave][lane][jl2 * 3 + 0];
            float cy = cbuf[wave][lane][jl2 * 3 + 1];
            float cz = cbuf[wave][lane][jl2 * 3 + 2];
            float4 acc = CL ? sample_cl(ws, g2, cx, cy, cz)
                            : sample_orig(feats, g2, cx, cy, cz);
            *(float4*)(outrow + g2 * 4) = acc;
        }
        __syncthreads();
    }
}

// Scalar tail for N % 256 points (none for the given sizes; keeps launch general).
template <bool CL>
__global__ void amg_tail(const float* __restrict__ x, const float* __restrict__ sc,
                         const float* __restrict__ tr, const float* __restrict__ ro,
                         const float* __restrict__ feats, const float* __restrict__ ws,
                         float* __restrict__ out, int start, int n) {
    int q = start + blockIdx.x * 256 + threadIdx.x;
    if (q >= n) return;
    float px = x[3 * q + 0], py = x[3 * q + 1], pz = x[3 * q + 2];
    for (int g = 0; g < NGRIDS; ++g) {
        float T[12];
        compute_transform(sc, tr, ro, g, T);
        float cx = T[0] * px + T[1] * py + T[2] * pz + T[9];
        float cy = T[3] * px + T[4] * py + T[5] * pz + T[10];
        float cz = T[6] * px + T[7] * py + T[8] * pz + T[11];
        float4 acc = CL ? sample_cl(ws, g, cx, cy, cz)
                        : sample_orig(feats, g, cx, cy, cz);
        *(float4*)(out + (size_t)q * (NGRIDS * NFEAT) + g * 4) = acc;
    }
}

extern "C" void kernel_launch(void* const* d_in, const int* in_sizes, int n_in,
                              void* d_out, int out_size, void* d_ws, size_t ws_size,
                              hipStream_t stream) {
    const float* x  = (const float*)d_in[0];
    const float* sc = (const float*)d_in[1];
    const float* tr = (const float*)d_in[2];
    const float* ro = (const float*)d_in[3];
    const float* fg = (const float*)d_in[4];
    float* out = (float*)d_out;
    float* ws  = (float*)d_ws;

    const int n = in_sizes[0] / 3;
    const size_t need = (size_t)NGRIDS * NFEAT * GCELLS * sizeof(float);
    const bool cl = (ws_size >= need);

    if (cl) {
        amg_repack<<<(NGRIDS * GCELLS) / 256, 256, 0, stream>>>(fg, ws);
    }

    const int nMain = (n / 256) * 256;
    if (nMain > 0) {
        if (cl) amg_main<true ><<<nMain / 256, 256, 0, stream>>>(x, sc, tr, ro, fg, ws, out);
        else    amg_main<false><<<nMain / 256, 256, 0, stream>>>(x, sc, tr, ro, fg, ws, out);
    }
    const int rem = n - nMain;
    if (rem > 0) {
        const int blocks = (rem + 255) / 256;
        if (cl) amg_tail<true ><<<blocks, 256, 0, stream>>>(x, sc, tr, ro, fg, ws, out, nMain, n);
        else    amg_tail<false><<<blocks, 256, 0, stream>>>(x, sc, tr, ro, fg, ws, out, nMain, n);
    }
}